// Sinkhorn_38800734552853
// MI455X (gfx1250) — compile-verified
//
#include <hip/hip_runtime.h>
#include <cmath>

typedef __attribute__((ext_vector_type(2))) float v2f;
typedef __attribute__((ext_vector_type(8))) float v8f;

#define BATCH 2
#define NPTS 4096
#define CS 8                                   // column chunks per row strip
#define COLS_PER_CHUNK (NPTS / CS)             // 512
#define TILES_PER_CHUNK (COLS_PER_CHUNK / 16)  // 32
#define STRIPS (NPTS / 16)                     // 256

// ---------------- precompute 0.5*||p||^2 for both clouds ----------------
__global__ void sq_norm_kernel(const float* __restrict__ X, const float* __restrict__ Y,
                               float* __restrict__ sqX, float* __restrict__ sqY) {
  int i = blockIdx.x * blockDim.x + threadIdx.x;
  if (i < BATCH * NPTS) {
    float x0 = X[3 * i], x1 = X[3 * i + 1], x2 = X[3 * i + 2];
    sqX[i] = 0.5f * (x0 * x0 + x1 * x1 + x2 * x2);
    float y0 = Y[3 * i], y1 = Y[3 * i + 1], y2 = Y[3 * i + 2];
    sqY[i] = 0.5f * (y0 * y0 + y1 * y1 + y2 * y2);
  }
}

// ---------------- per-column dual bias: g_j = logw + (pot_j - sqCol_j)/eps ----------------
__global__ void prep_g_kernel(const float* __restrict__ sqCol, const float* __restrict__ pot,
                              float logw, float inv_eps, float* __restrict__ g) {
  int i = blockIdx.x * blockDim.x + threadIdx.x;
  if (i >= BATCH * NPTS) return;
  float v = -sqCol[i];
  if (pot) v += pot[i];
  g[i] = logw + v * inv_eps;
}

// ---------------- softmin stage 1: WMMA-tiled online logsumexp ----------------
// f_i = sqX_i - eps * log sum_j exp(g_j + y_j.(x_i/eps))
// Operand roles transposed: A = y tile (M dim = j), B = x strip (N dim = i).
// D layout => lane holds 8 consecutive j's of ONE row i: single (m,s) per lane,
// shared tile max => 9 exps per 8 elements.
__global__ __launch_bounds__(256) void softmin_stage1(
    const float* __restrict__ Prow, const float* __restrict__ Pcol,
    const float* __restrict__ g, float inv_eps,
    float* __restrict__ pm, float* __restrict__ ps) {
  const int gt = blockIdx.x * blockDim.x + threadIdx.x;
  const int wave = gt >> 5;
  const int lane = gt & 31;
  const int chunk = wave & (CS - 1);
  const int strip = (wave >> 3) & (STRIPS - 1);
  const int b = wave >> 11;  // 3 + 8 bits
  const int m16 = lane & 15;
  const int hi = lane >> 4;
  const size_t bbase = (size_t)b * NPTS;
  const int i0 = strip * 16;

  // B (4x16 f32): column n = output row i; lanes 0-15 hold K=0,1; 16-31 hold K=2,3.
  const float* xr = Prow + (bbase + i0 + m16) * 3;
  v2f Bx;
  Bx.x = (hi ? xr[2] : xr[0]) * inv_eps;
  Bx.y = (hi ? 0.0f : xr[1]) * inv_eps;

  float m = -INFINITY, s = 0.0f;
  const int jbase = chunk * COLS_PER_CHUNK;
#pragma unroll 2
  for (int t = 0; t < TILES_PER_CHUNK; ++t) {
    const int j0 = jbase + t * 16;
    // A (16x4 f32): row M = column index j; lanes 0-15 hold K=0,1; 16-31 hold K=2,3.
    const float* yc = Pcol + (bbase + j0 + m16) * 3;
    v2f A;
    A.x = hi ? yc[2] : yc[0];
    A.y = hi ? 0.0f : yc[1];
    // this lane's 8 j's are j0+8*hi .. j0+8*hi+7 (broadcast across half-wave)
    const float* gp = g + bbase + j0 + 8 * hi;
    const float4 gl0 = *(const float4*)(gp);
    const float4 gl1 = *(const float4*)(gp + 4);

    v8f c = {};
    v8f d = __builtin_amdgcn_wmma_f32_16x16x4_f32(false, A, false, Bx,
                                                  (short)0, c, false, false);
    // lane (l, vgpr r): element (j = j0 + 8*hi + r, i = i0 + m16)
    float a0 = d[0] + gl0.x, a1 = d[1] + gl0.y, a2 = d[2] + gl0.z, a3 = d[3] + gl0.w;
    float a4 = d[4] + gl1.x, a5 = d[5] + gl1.y, a6 = d[6] + gl1.z, a7 = d[7] + gl1.w;
    float tm = fmaxf(fmaxf(fmaxf(a0, a1), fmaxf(a2, a3)),
                     fmaxf(fmaxf(a4, a5), fmaxf(a6, a7)));
    float nm = fmaxf(m, tm);
    float sum = __expf(a0 - nm) + __expf(a1 - nm) + __expf(a2 - nm) + __expf(a3 - nm) +
                __expf(a4 - nm) + __expf(a5 - nm) + __expf(a6 - nm) + __expf(a7 - nm);
    s = s * __expf(m - nm) + sum;
    m = nm;
  }

  // lane l and l+16 hold the two j-halves of the same row i: one combine.
  {
    float om = __shfl_xor(m, 16, 32);
    float os = __shfl_xor(s, 16, 32);
    float nm = fmaxf(m, om);
    s = s * __expf(m - nm) + os * __expf(om - nm);
    m = nm;
  }
  if (!hi) {
    size_t idx = (bbase + i0 + m16) * CS + chunk;
    pm[idx] = m;
    ps[idx] = s;
  }
}

// ---------------- softmin stage 2: merge chunk partials, finalize ----------------
__global__ void softmin_stage2(const float* __restrict__ pm, const float* __restrict__ ps,
                               const float* __restrict__ sqRow, float eps,
                               float* __restrict__ out) {
  int i = blockIdx.x * blockDim.x + threadIdx.x;
  if (i >= BATCH * NPTS) return;
  float m = -INFINITY;
#pragma unroll
  for (int c = 0; c < CS; ++c) m = fmaxf(m, pm[(size_t)i * CS + c]);
  float s = 0.0f;
#pragma unroll
  for (int c = 0; c < CS; ++c)
    s += ps[(size_t)i * CS + c] * __expf(pm[(size_t)i * CS + c] - m);
  out[i] = sqRow[i] - eps * (m + __logf(s));
}

// ---------------- symmetrized damping updates ----------------
__global__ void avg4_kernel(float* __restrict__ ay, const float* __restrict__ tay,
                            float* __restrict__ bx, const float* __restrict__ tbx,
                            float* __restrict__ ax, const float* __restrict__ tax,
                            float* __restrict__ by, const float* __restrict__ tby) {
  int i = blockIdx.x * blockDim.x + threadIdx.x;
  if (i >= BATCH * NPTS) return;
  ay[i] = 0.5f * (ay[i] + tay[i]);
  bx[i] = 0.5f * (bx[i] + tbx[i]);
  ax[i] = 0.5f * (ax[i] + tax[i]);
  by[i] = 0.5f * (by[i] + tby[i]);
}

__global__ void zero_out_kernel(float* out) { out[0] = 0.0f; }

__global__ void final_reduce_kernel(const float* __restrict__ tbx, const float* __restrict__ tax,
                                    const float* __restrict__ tay, const float* __restrict__ tby,
                                    float* __restrict__ out) {
  __shared__ float sm[256];
  int i = blockIdx.x * blockDim.x + threadIdx.x;
  float v = 0.0f;
  if (i < BATCH * NPTS)
    v = (tbx[i] - tax[i] + tay[i] - tby[i]) * (1.0f / (float)NPTS);
  sm[threadIdx.x] = v;
  __syncthreads();
  for (int s = 128; s > 0; s >>= 1) {
    if ((int)threadIdx.x < s) sm[threadIdx.x] += sm[threadIdx.x + s];
    __syncthreads();
  }
  if (threadIdx.x == 0) atomicAdd(out, sm[0]);
}

// ---------------- host orchestration ----------------
extern "C" void kernel_launch(void* const* d_in, const int* in_sizes, int n_in,
                              void* d_out, int out_size, void* d_ws, size_t ws_size,
                              hipStream_t stream) {
  const float* X = (const float*)d_in[0];  // true_data [2,4096,3]
  const float* Y = (const float*)d_in[1];  // particles [2,4096,3]
  float* out = (float*)d_out;
  float* ws = (float*)d_ws;

  const int BN = BATCH * NPTS;
  float* sqX = ws;
  float* sqY = sqX + BN;
  float* ax = sqY + BN;
  float* by = ax + BN;
  float* bx = by + BN;
  float* ay = bx + BN;
  float* tax = ay + BN;
  float* tby = tax + BN;
  float* tbx = tby + BN;
  float* tay = tbx + BN;
  float* gbuf = tay + BN;               // BN
  float* pm = gbuf + BN;                // BN * CS
  float* ps = pm + (size_t)BN * CS;     // BN * CS

  const float logw = -logf((float)NPTS);  // a_log == b_log (N == M)

  // geomloss epsilon schedule: [diam^2] + exp(arange(2ln4, 2ln0.05, 2ln0.5)) + [blur^2]
  double eps_list[16];
  int n_eps = 0;
  eps_list[n_eps++] = 16.0;
  for (double e = 2.0 * log(4.0); e > 2.0 * log(0.05); e += 2.0 * log(0.5))
    eps_list[n_eps++] = exp(e);
  eps_list[n_eps++] = 0.05 * 0.05;

  dim3 blk(256);
  dim3 gvec((BN + 255) / 256);
  const int s1_blocks = (BATCH * STRIPS * CS * 32) / 256;  // 512 blocks of 8 waves

  sq_norm_kernel<<<gvec, blk, 0, stream>>>(X, Y, sqX, sqY);

  auto softmin = [&](const float* Prow, const float* Pcol, const float* sqR,
                     const float* sqC, const float* pot, double eps, float* outv) {
    float inv_eps = (float)(1.0 / eps);
    prep_g_kernel<<<gvec, blk, 0, stream>>>(sqC, pot, logw, inv_eps, gbuf);
    softmin_stage1<<<dim3(s1_blocks), blk, 0, stream>>>(Prow, Pcol, gbuf, inv_eps,
                                                        pm, ps);
    softmin_stage2<<<gvec, blk, 0, stream>>>(pm, ps, sqR, (float)eps, outv);
  };

  // init (h = log weights only)
  double eps0 = eps_list[0];
  softmin(X, X, sqX, sqX, nullptr, eps0, ax);  // a_x = softmin(C_xx, a_log)
  softmin(Y, Y, sqY, sqY, nullptr, eps0, by);  // b_y = softmin(C_yy, b_log)
  softmin(X, Y, sqX, sqY, nullptr, eps0, bx);  // b_x = softmin(C_xy, b_log)
  softmin(Y, X, sqY, sqX, nullptr, eps0, ay);  // a_y = softmin(C_yx, a_log)

  // eps-scaling descent with symmetrized updates
  for (int it = 0; it < n_eps; ++it) {
    double eps = eps_list[it];
    softmin(Y, X, sqY, sqX, bx, eps, tay);  // at_y = softmin(C_yx, a_log + b_x/eps)
    softmin(X, Y, sqX, sqY, ay, eps, tbx);  // bt_x = softmin(C_xy, b_log + a_y/eps)
    softmin(X, X, sqX, sqX, ax, eps, tax);  // at_x = softmin(C_xx, a_log + a_x/eps)
    softmin(Y, Y, sqY, sqY, by, eps, tby);  // bt_y = softmin(C_yy, b_log + b_y/eps)
    avg4_kernel<<<gvec, blk, 0, stream>>>(ay, tay, bx, tbx, ax, tax, by, tby);
  }

  // last extrapolation (reads old potentials, writes temps)
  double epsL = eps_list[n_eps - 1];
  softmin(Y, X, sqY, sqX, bx, epsL, tay);  // new a_y
  softmin(X, Y, sqX, sqY, ay, epsL, tbx);  // new b_x
  softmin(X, X, sqX, sqX, ax, epsL, tax);  // new a_x
  softmin(Y, Y, sqY, sqY, by, epsL, tby);  // new b_y

  zero_out_kernel<<<1, 1, 0, stream>>>(out);
  final_reduce_kernel<<<gvec, blk, 0, stream>>>(tbx, tax, tay, tby, out);
}